// MultiPathChannel_30889404792870
// MI455X (gfx1250) — compile-verified
//
#include <hip/hip_runtime.h>
#include <hip/hip_bf16.h>
#include <stdint.h>

typedef float v2f __attribute__((ext_vector_type(2)));
typedef float v8f __attribute__((ext_vector_type(8)));

#define B_DIM   128
#define T_DIM   8192
#define TAPS    32
#define TILE_T  128          // t outputs per workgroup
#define THREADS 256          // 8 wave32
#define ROW_PITCH 34         // padded LDS row pitch (floats) -> conflict-free b64 reads
#define XWIN    161          // staged x window: t0-32 .. t0+127 (+1 pad)

__device__ __forceinline__ uint32_t hash_u32(uint32_t x) {
    x ^= x >> 16; x *= 0x85EBCA6Bu;
    x ^= x >> 13; x *= 0xC2B2AE35u;
    x ^= x >> 16;
    return x;
}

__global__ __launch_bounds__(THREADS, 2)
void mpc_kernel(const float* __restrict__ x_real,
                const float* __restrict__ x_imag,
                const float* __restrict__ h_real,
                const float* __restrict__ h_imag,
                const float* __restrict__ npdb,
                float* __restrict__ out)
{
    __shared__ __align__(16) float sh_hr[TILE_T * ROW_PITCH];
    __shared__ __align__(16) float sh_hi[TILE_T * ROW_PITCH];
    __shared__ float sh_xr[XWIN];
    __shared__ float sh_xi[XWIN];

    const int b   = blockIdx.y;
    const int t0  = blockIdx.x * TILE_T;
    const int tid = threadIdx.x;

    // ---------- Stage h tile (TILE_T x TAPS, re+im) into LDS: async copy ----------
    const float* gr = h_real + ((size_t)b * T_DIM + t0) * TAPS;
    const float* gi = h_imag + ((size_t)b * T_DIM + t0) * TAPS;
    #pragma unroll
    for (int j = 0; j < (TILE_T * TAPS) / (THREADS * 4); ++j) {   // 4 iterations
        int e4  = j * THREADS + tid;       // float4 index within tile
        int row = e4 >> 3;                 // 8 float4 per 32-tap row
        int col = (e4 & 7) * 4;
        unsigned ldsr = (unsigned)(uintptr_t)&sh_hr[row * ROW_PITCH + col];
        unsigned ldsi = (unsigned)(uintptr_t)&sh_hi[row * ROW_PITCH + col];
        const float* ga = gr + (size_t)e4 * 4;
        const float* gb = gi + (size_t)e4 * 4;
        asm volatile("global_load_async_to_lds_b128 %0, %1, off"
                     :: "v"(ldsr), "v"(ga) : "memory");
        asm volatile("global_load_async_to_lds_b128 %0, %1, off"
                     :: "v"(ldsi), "v"(gb) : "memory");
    }

    // ---------- Stage x window [t0-32, t0+127] into LDS (zero-padded) ----------
    if (tid < 160) {
        int t  = t0 - 32 + tid;
        int tc = t < 0 ? 0 : t;            // clamped load + select (no divergent load)
        float fr = x_real[(size_t)b * T_DIM + tc];
        float fi = x_imag[(size_t)b * T_DIM + tc];
        sh_xr[tid] = (t < 0) ? 0.0f : fr;
        sh_xi[tid] = (t < 0) ? 0.0f : fi;
    }

    asm volatile("s_wait_asynccnt 0" ::: "memory");
    __syncthreads();

    // ---------- Compute: 16 complex outputs per wave ----------
    const int wv   = tid >> 5;
    const int lane = tid & 31;
    const int m    = lane & 15;            // output row within wave tile (A-matrix M)
    const int kb   = (lane >> 4) * 2;      // tap sub-offset per WMMA A layout (K pairs)
    const int tw   = t0 + wv * 16;
    const int rowL = wv * 16 + m;
    const int xbase = wv * 16 + m + 32;    // LDS x index for tap 0

    v8f accR = {};
    v8f accI = {};
    const v2f onesB = {1.0f, 1.0f};        // B = all-ones -> D[m,*] = row-sum over K

    #pragma unroll
    for (int k0 = 0; k0 < TAPS; k0 += 4) {
        const int k   = k0 + kb;           // this lane's taps: k, k+1
        const int off = rowL * ROW_PITCH + k;
        float hr0 = sh_hr[off];
        float hr1 = sh_hr[off + 1];
        float hi0 = sh_hi[off];
        float hi1 = sh_hi[off + 1];

        const int i0 = xbase - k;          // x sample index (LDS) for tap k
        float xr0 = sh_xr[i0];
        float xi0 = sh_xi[i0];
        float xr1 = sh_xr[i0 - 1];         // tap k+1
        float xi1 = sh_xi[i0 - 1];

        // complex products (VALU); WMMA performs the K-reduction
        v2f aR = { hr0 * xr0 - hi0 * xi0, hr1 * xr1 - hi1 * xi1 };
        v2f aI = { hr0 * xi0 + hi0 * xr0, hr1 * xi1 + hi1 * xr1 };

        accR = __builtin_amdgcn_wmma_f32_16x16x4_f32(
                   false, aR, false, onesB, (short)0, accR, false, false);
        accI = __builtin_amdgcn_wmma_f32_16x16x4_f32(
                   false, aI, false, onesB, (short)0, accI, false, false);
    }

    // ---------- Noise + store ----------
    // D layout: vgpr j, lanes 0-15 -> M=j ; lanes 16-31 -> M=8+j (all columns equal)
    if ((lane & 8) == 0) {
        const int j  = lane & 7;
        const int mo = j + ((lane >> 4) << 3);
        const int t  = tw + mo;
        float yrv = accR[j];
        float yiv = accI[j];

        // sigma = sqrt(0.5 * 10^(dB/10))
        float np  = npdb[b];
        float sd  = sqrtf(0.5f * exp2f(np * 0.3321928094887362f));

        // deterministic per-element gaussian (hash + Box-Muller)
        uint32_t idx = (uint32_t)(b * T_DIM + t);
        uint32_t h1 = hash_u32(idx * 2654435761u + 0x9E3779B9u);
        uint32_t h2 = hash_u32((idx ^ 0xDEADBEEFu) * 0x27D4EB2Du + 0x165667B1u);
        float u1 = (float)(h1 >> 8) * (1.0f / 16777216.0f) + 1.0e-7f;
        float u2 = (float)(h2 >> 8) * (1.0f / 16777216.0f);
        float r  = sqrtf(-2.0f * __logf(u1));
        float n0 = r * __cosf(6.28318530718f * u2);
        float n1 = r * __sinf(6.28318530718f * u2);

        size_t o = ((size_t)b * T_DIM + t) * 2;
        out[o + 0] = yrv + sd * n0;
        out[o + 1] = yiv + sd * n1;
    }
}

extern "C" void kernel_launch(void* const* d_in, const int* in_sizes, int n_in,
                              void* d_out, int out_size, void* d_ws, size_t ws_size,
                              hipStream_t stream) {
    const float* x_real = (const float*)d_in[0];
    const float* x_imag = (const float*)d_in[1];
    const float* h_real = (const float*)d_in[2];
    const float* h_imag = (const float*)d_in[3];
    const float* npdb   = (const float*)d_in[4];
    float* out = (float*)d_out;

    dim3 grid(T_DIM / TILE_T, B_DIM, 1);   // 64 x 128 blocks
    mpc_kernel<<<grid, THREADS, 0, stream>>>(x_real, x_imag, h_real, h_imag, npdb, out);
}